// TVL1OF_14551349199636
// MI455X (gfx1250) — compile-verified
//
#include <hip/hip_runtime.h>

#define H_    256
#define W_    256
#define NB_   12          // B*C image pairs
#define TH_   32
#define TW_   32
#define NTX_  8
#define NTY_  8
#define PW_   264         // padded plane pitch (floats), 32B-aligned rows
#define PH_   260         // allocated padded rows (258 used, spare rows stay zero)
#define PP_   (PH_*PW_)   // 68640 floats per padded plane
#define EW_   (TW_+1)     // 33: extended u_new region
#define EH_   (TH_+1)
#define PTW_  (TW_+3)     // 35: p tile with +/-1 halo around extended region
#define PTH_  (TH_+3)
#define EPS_  1e-8f
#define NUM_ITER_ 20

// ---------------- TDM (Tensor Data Mover) tile load -------------------------
#if defined(__has_builtin)
#if __has_builtin(__builtin_amdgcn_tensor_load_to_lds) && __has_builtin(__builtin_amdgcn_s_wait_tensorcnt)
#define HAVE_TDM 1
#endif
#endif
#ifndef HAVE_TDM
#define HAVE_TDM 0
#endif

#if HAVE_TDM
typedef __attribute__((ext_vector_type(4))) unsigned int tdm_u4;
typedef __attribute__((ext_vector_type(8))) int          tdm_i8;
typedef __attribute__((ext_vector_type(4))) int          tdm_i4;

// DMA a PTH_ x PTW_ f32 tile (row pitch PW_ elements) from global -> LDS.
// All addresses stay inside allocated, zero-initialized plane memory, so we
// do not rely on TDM OOB semantics at image borders.
__device__ __forceinline__ void tdm_tile(const float* g, float* lds) {
  unsigned long long ga = (unsigned long long)(size_t)g;
  unsigned lo = (unsigned)((size_t)lds & 0xFFFFFFFFull); // LDS byte offset (addr[31:0])
  tdm_u4 g0;
  g0[0] = 1u;                                            // count=1, user descriptor
  g0[1] = lo;                                            // lds_addr
  g0[2] = (unsigned)(ga & 0xFFFFFFFFull);                // global_addr[31:0]
  g0[3] = (unsigned)((ga >> 32) & 0x01FFFFFFull) | (2u << 30); // addr[56:32] | type=2
  tdm_i8 g1;
  g1[0] = (int)(2u << 16);                               // data_size = 4 bytes
  g1[1] = (int)(((unsigned)PW_ & 0xFFFFu) << 16);        // tensor_dim0[15:0] @ bits63:48
  g1[2] = (int)(((unsigned)PH_ & 0xFFFFu) << 16);        // dim0 hi16=0 | tensor_dim1 lo16
  g1[3] = (int)((unsigned)PTW_ << 16);                   // dim1 hi16=0 | tile_dim0
  g1[4] = (int)PTH_;                                     // tile_dim1 | tile_dim2=0
  g1[5] = (int)PW_;                                      // tensor_dim0_stride[31:0]
  g1[6] = 0;
  g1[7] = 0;
  tdm_i4 gz4 = {0, 0, 0, 0};
  tdm_i8 gz8 = {0, 0, 0, 0, 0, 0, 0, 0};
  // clang-23 / therock-10.0 lane: 6-arg form (g0, g1, g2, g3, g_extra, cpol)
  __builtin_amdgcn_tensor_load_to_lds(g0, g1, gz4, gz4, gz8, 0);
}
#endif

// ---------------- helpers ---------------------------------------------------
__device__ __forceinline__ float rd_img(const float* p, int y, int x) {
  return (y >= 0 && y < H_ && x >= 0 && x < W_) ? p[y * W_ + x] : 0.f;
}

// ---------------- kernels ---------------------------------------------------
__global__ void tv_zero(float* __restrict__ p, int n4) {
  int i = blockIdx.x * blockDim.x + threadIdx.x;
  if (i < n4) ((float4*)p)[i] = make_float4(0.f, 0.f, 0.f, 0.f);
}

// rho_c = I1 - I0 ; Sobel/6 gradients of I1 (zero padded) ; planes unpadded.
__global__ void tv_pre(const float* __restrict__ xin, float* __restrict__ rho,
                       float* __restrict__ gxo, float* __restrict__ gyo) {
  int i = blockIdx.x * blockDim.x + threadIdx.x;
  if (i >= NB_ * H_ * W_) return;
  int x = i % W_, y = (i / W_) % H_, n = i / (H_ * W_);
  const float* I0 = xin + (size_t)n * H_ * W_;
  const float* I1 = xin + (size_t)(NB_ + n) * H_ * W_;
  rho[i] = I1[y * W_ + x] - I0[y * W_ + x];
  float a00 = rd_img(I1, y - 1, x - 1), a01 = rd_img(I1, y - 1, x), a02 = rd_img(I1, y - 1, x + 1);
  float a10 = rd_img(I1, y,     x - 1),                              a12 = rd_img(I1, y,     x + 1);
  float a20 = rd_img(I1, y + 1, x - 1), a21 = rd_img(I1, y + 1, x), a22 = rd_img(I1, y + 1, x + 1);
  gxo[i] = ((a02 + 2.f * a12 + a22) - (a00 + 2.f * a10 + a20)) * (1.f / 6.f);
  gyo[i] = ((a20 + 2.f * a21 + a22) - (a00 + 2.f * a01 + a02)) * (1.f / 6.f);
}

// One fused TV-L1 iteration per launch (halo recomputation of u_new).
__global__ __launch_bounds__(256) void tv_iter(
    const float* __restrict__ uS, const float* __restrict__ p1S, const float* __restrict__ p2S,
    float* __restrict__ uD, float* __restrict__ p1D, float* __restrict__ p2D,
    const float* __restrict__ rho, const float* __restrict__ gxp, const float* __restrict__ gyp,
    const float* __restrict__ lam_p, const float* __restrict__ tau_p, const float* __restrict__ th_p,
    const float* __restrict__ wx_p, const float* __restrict__ wy_p) {
  __shared__ float sp[4][PTH_ * PTW_];   // p1x,p1y,p2x,p2y tiles (with halo)
  __shared__ float su[2][EH_ * EW_];     // u_new extended region

  const int b = blockIdx.x;
  const int n = b / (NTX_ * NTY_);
  const int t = b % (NTX_ * NTY_);
  const int y0 = (t / NTX_) * TH_;
  const int x0 = (t % NTX_) * TW_;
  const int tid = threadIdx.x;

  const float theta = th_p[0];
  const float tl = theta * lam_p[0];
  const float tt = tau_p[0] / theta;
  const float wx0 = wx_p[0], wx1 = wx_p[1], wx2 = wx_p[2];
  const float wy0 = wy_p[0], wy1 = wy_p[1], wy2 = wy_p[2];

  const size_t pb = (size_t)(n * 2) * PP_;
  const float* us0  = uS  + pb;       const float* us1  = uS  + pb + PP_;
  const float* p1xs = p1S + pb;       const float* p1ys = p1S + pb + PP_;
  const float* p2xs = p2S + pb;       const float* p2ys = p2S + pb + PP_;
  const int toff = y0 * PW_ + x0;     // padded coord of global (y0-1, x0-1)

  // ---- stage p tiles into LDS ----
#if HAVE_TDM
  if (tid < 32u) {                    // one wave issues the 4 DMAs
    tdm_tile(p1xs + toff, &sp[0][0]);
    tdm_tile(p1ys + toff, &sp[1][0]);
    tdm_tile(p2xs + toff, &sp[2][0]);
    tdm_tile(p2ys + toff, &sp[3][0]);
    __builtin_amdgcn_s_wait_tensorcnt(0);
  }
#else
  for (int i = tid; i < PTH_ * PTW_; i += 256) {
    int r = i / PTW_, c = i % PTW_;
    int g = (y0 + r) * PW_ + (x0 + c);
    sp[0][i] = p1xs[g]; sp[1][i] = p1ys[g];
    sp[2][i] = p2xs[g]; sp[3][i] = p2ys[g];
  }
#endif
  __syncthreads();

  // ---- phase 1: u_new on extended (TH+1)x(TW+1) region ----
  const float* rhn = rho + (size_t)n * H_ * W_;
  const float* gxn = gxp + (size_t)n * H_ * W_;
  const float* gyn = gyp + (size_t)n * H_ * W_;
  for (int i = tid; i < EH_ * EW_; i += 256) {
    int ey = i / EW_, ex = i % EW_;
    int yy = y0 + ey, xx = x0 + ex;       // may equal H_/W_ at image edges
    float un0 = 0.f, un1 = 0.f;
    if (yy < H_ && xx < W_) {             // out-of-image extended points stay 0 (zero-pad)
      int pix = yy * W_ + xx;
      float gxv = gxn[pix], gyv = gyn[pix];
      int pad = (yy + 1) * PW_ + (xx + 1);
      float u0 = us0[pad], u1 = us1[pad];
      float r = rhn[pix] + gxv * u0 + gyv * u1;
      float ng = gxv * gxv + gyv * gyv + EPS_;
      float v0, v1;
      if (fabsf(r) < tl * ng) {
        float d = r / ng;
        v0 = u0 - d * gxv; v1 = u1 - d * gyv;
      } else {
        float s = (r > 0.f) ? 1.f : ((r < 0.f) ? -1.f : 0.f);
        v0 = u0 - tl * s * gxv; v1 = u1 - tl * s * gyv;
      }
      int tc = (ey + 1) * PTW_ + (ex + 1);
      float div1 = wx0 * sp[0][tc - 1]    + wx1 * sp[0][tc] + wx2 * sp[0][tc + 1]
                 + wy0 * sp[1][tc - PTW_] + wy1 * sp[1][tc] + wy2 * sp[1][tc + PTW_];
      float div2 = wx0 * sp[2][tc - 1]    + wx1 * sp[2][tc] + wx2 * sp[2][tc + 1]
                 + wy0 * sp[3][tc - PTW_] + wy1 * sp[3][tc] + wy2 * sp[3][tc + PTW_];
      un0 = v0 + theta * div1;
      un1 = v1 + theta * div2;
    }
    su[0][i] = un0;
    su[1][i] = un1;
  }
  __syncthreads();

  // ---- phase 2: p update on interior tile + writeback ----
  float* ud0  = uD  + pb;  float* ud1  = uD  + pb + PP_;
  float* p1xd = p1D + pb;  float* p1yd = p1D + pb + PP_;
  float* p2xd = p2D + pb;  float* p2yd = p2D + pb + PP_;
  for (int i = tid; i < TH_ * TW_; i += 256) {
    int iy = i / TW_, ix = i % TW_;
    int e = iy * EW_ + ix;
    float u0c = su[0][e], u0r = su[0][e + 1], u0d = su[0][e + EW_];
    float u1c = su[1][e], u1r = su[1][e + 1], u1d = su[1][e + EW_];
    float g1x = u0r - u0c, g1y = u0d - u0c;   // forward diffs (zero-pad via extended=0)
    float g2x = u1r - u1c, g2y = u1d - u1c;
    float inv1 = 1.f / (1.f + tt * (fabsf(g1x) + fabsf(g1y)));
    float inv2 = 1.f / (1.f + tt * (fabsf(g2x) + fabsf(g2y)));
    int tc = (iy + 1) * PTW_ + (ix + 1);
    int po = (y0 + iy + 1) * PW_ + (x0 + ix + 1);
    p1xd[po] = (sp[0][tc] + tt * g1x) * inv1;
    p1yd[po] = (sp[1][tc] + tt * g1y) * inv1;
    p2xd[po] = (sp[2][tc] + tt * g2x) * inv2;
    p2yd[po] = (sp[3][tc] + tt * g2y) * inv2;
    ud0[po] = u0c;
    ud1[po] = u1c;
  }
}

__global__ void tv_out(const float* __restrict__ u, float* __restrict__ out) {
  int i = blockIdx.x * blockDim.x + threadIdx.x;
  if (i >= NB_ * 2 * H_ * W_) return;
  int x = i % W_, y = (i / W_) % H_;
  int c = (i / (H_ * W_)) % 2, n = i / (2 * H_ * W_);
  out[i] = u[(size_t)(n * 2 + c) * PP_ + (y + 1) * PW_ + (x + 1)];
}

// ---------------- host ------------------------------------------------------
extern "C" void kernel_launch(void* const* d_in, const int* in_sizes, int n_in,
                              void* d_out, int out_size, void* d_ws, size_t ws_size,
                              hipStream_t stream) {
  const float* xin   = (const float*)d_in[0];
  const float* lam   = (const float*)d_in[1];
  const float* tau   = (const float*)d_in[2];
  const float* theta = (const float*)d_in[3];
  const float* wx    = (const float*)d_in[4];
  const float* wy    = (const float*)d_in[5];
  float* out = (float*)d_out;
  float* ws  = (float*)d_ws;

  // workspace layout: 6 padded plane sets (uA,uB,p1A,p1B,p2A,p2B) + rho,gx,gy
  float* uP[2];  float* p1P[2];  float* p2P[2];
  uP[0]  = ws;                         uP[1]  = uP[0]  + (size_t)24 * PP_;
  p1P[0] = uP[1]  + (size_t)24 * PP_;  p1P[1] = p1P[0] + (size_t)24 * PP_;
  p2P[0] = p1P[1] + (size_t)24 * PP_;  p2P[1] = p2P[0] + (size_t)24 * PP_;
  float* rho = p2P[1] + (size_t)24 * PP_;
  float* gx  = rho + (size_t)NB_ * H_ * W_;
  float* gy  = gx  + (size_t)NB_ * H_ * W_;

  const int zeroN4 = (144 * PP_) / 4;            // zero all padded planes (borders too)
  tv_zero<<<(zeroN4 + 255) / 256, 256, 0, stream>>>(ws, zeroN4);

  const int npix = NB_ * H_ * W_;
  tv_pre<<<(npix + 255) / 256, 256, 0, stream>>>(xin, rho, gx, gy);

  int s = 0;
  for (int it = 0; it < NUM_ITER_; ++it) {
    int d = 1 - s;
    tv_iter<<<NB_ * NTY_ * NTX_, 256, 0, stream>>>(
        uP[s], p1P[s], p2P[s], uP[d], p1P[d], p2P[d],
        rho, gx, gy, lam, tau, theta, wx, wy);
    s = d;
  }
  // NUM_ITER_ even -> final u is in set 0 (s == 0 here)
  tv_out<<<(2 * npix + 255) / 256, 256, 0, stream>>>(uP[s], out);
}